// BSLClassifierV0_79663053406253
// MI455X (gfx1250) — compile-verified
//
#include <hip/hip_runtime.h>

typedef __attribute__((ext_vector_type(2))) float v2f;
typedef __attribute__((ext_vector_type(8))) float v8f;

#define NB_B 32768
#define NB_C 1000
#define NF4  250            // 1000 floats / 4 per float4

// workspace layout (bytes)
#define WS_HIST 0           // uint  [1000]
#define WS_FREQ 4096        // float [1000]
#define WS_LOGF 8192        // float [1000]
#define WS_PART 12288       // float [MAIN_BLOCKS]

#define MAIN_BLOCKS 2048
#define MAIN_THREADS 256
#define WAVES_PER_BLOCK (MAIN_THREADS / 32)

__global__ void bsl_zero_hist(unsigned* __restrict__ hist) {
    int i = blockIdx.x * blockDim.x + threadIdx.x;
    if (i < NB_C) hist[i] = 0u;
}

__global__ void bsl_hist(const int* __restrict__ target, unsigned* __restrict__ hist) {
    __shared__ unsigned lh[NB_C];
    for (int i = threadIdx.x; i < NB_C; i += blockDim.x) lh[i] = 0u;
    __syncthreads();
    int idx = blockIdx.x * blockDim.x + threadIdx.x;
    int stride = gridDim.x * blockDim.x;
    for (int i = idx; i < NB_B; i += stride) {
        atomicAdd(&lh[target[i]], 1u);
    }
    __syncthreads();
    for (int i = threadIdx.x; i < NB_C; i += blockDim.x) {
        unsigned v = lh[i];
        if (v) atomicAdd(&hist[i], v);
    }
}

__global__ void bsl_freq(const unsigned* __restrict__ hist,
                         float* __restrict__ freqf,
                         float* __restrict__ logfreq) {
    int i = blockIdx.x * blockDim.x + threadIdx.x;
    if (i < NB_C) {
        float f = (float)hist[i];
        freqf[i]   = f;
        logfreq[i] = __logf(f);
    }
}

__global__ __launch_bounds__(MAIN_THREADS)
void bsl_main(const float* __restrict__ pred,
              const int*   __restrict__ target,
              const float* __restrict__ freqf,
              const float* __restrict__ logfreq,
              float* __restrict__ partials) {
    const int lane  = threadIdx.x & 31;
    const int wid   = threadIdx.x >> 5;
    const int gwave = blockIdx.x * WAVES_PER_BLOCK + wid;
    const int nwave = gridDim.x * WAVES_PER_BLOCK;

    const float4* __restrict__ freq4 = (const float4*)freqf;

    float wave_acc = 0.0f;

    for (int row = gwave; row < NB_B; row += nwave) {
        const float4* __restrict__ p4 = (const float4*)(pred + (size_t)row * NB_C);

        // Two per-lane partials -> fill the 16x4 f32 A-matrix layout
        // (lanes 0-15: K=0,1 ; lanes 16-31: K=2,3).
        float acc0 = 0.0f, acc1 = 0.0f;
#pragma unroll
        for (int k = 0; k < 8; ++k) {
            int i4 = k * 32 + lane;
            if (i4 < NF4) {
                float4 p = p4[i4];
                float4 w = freq4[i4];
                float e = fmaf(__expf(p.x), w.x,
                          fmaf(__expf(p.y), w.y,
                          fmaf(__expf(p.z), w.z,
                               __expf(p.w) * w.w)));
                if (k & 1) acc1 += e; else acc0 += e;
            }
        }

        // Wave-wide reduction on the matrix pipe:
        // D = A(16x4 partials) x B(4x16 ones) ; D[m,n] = sum_k A[m,k]
        v2f a; a.x = acc0; a.y = acc1;
        v2f b; b.x = 1.0f; b.y = 1.0f;
        v8f c = {};
        v8f d = __builtin_amdgcn_wmma_f32_16x16x4_f32(
                    /*neg_a=*/false, a, /*neg_b=*/false, b,
                    /*c_mod=*/(short)0, c, /*reuse_a=*/false, /*reuse_b=*/false);

        // lanes 0-15 hold M=0..7 sums in d[0..7]; lanes 16-31 hold M=8..15.
        float s = d[0] + d[1] + d[2] + d[3] + d[4] + d[5] + d[6] + d[7];
        float rowsum = s + __shfl_xor(s, 16, 32);

        if (lane == 0) {
            int t = target[row];
            float g = pred[(size_t)row * NB_C + t];
            wave_acc += __logf(rowsum) - g - logfreq[t];
        }
    }

    __shared__ float wsum[WAVES_PER_BLOCK];
    if (lane == 0) wsum[wid] = wave_acc;
    __syncthreads();
    if (threadIdx.x == 0) {
        float s = 0.0f;
#pragma unroll
        for (int i = 0; i < WAVES_PER_BLOCK; ++i) s += wsum[i];
        partials[blockIdx.x] = s;
    }
}

__global__ void bsl_final(const float* __restrict__ partials, float* __restrict__ out) {
    __shared__ float sm[1024];
    int t = threadIdx.x;
    sm[t] = partials[t] + partials[t + 1024];
    __syncthreads();
    for (int off = 512; off > 0; off >>= 1) {
        if (t < off) sm[t] += sm[t + off];
        __syncthreads();
    }
    if (t == 0) out[0] = sm[0] / (float)NB_B;
}

extern "C" void kernel_launch(void* const* d_in, const int* in_sizes, int n_in,
                              void* d_out, int out_size, void* d_ws, size_t ws_size,
                              hipStream_t stream) {
    const float* pred   = (const float*)d_in[0];
    const int*   target = (const int*)d_in[1];

    char* ws = (char*)d_ws;
    unsigned* hist    = (unsigned*)(ws + WS_HIST);
    float*    freqf   = (float*)(ws + WS_FREQ);
    float*    logfreq = (float*)(ws + WS_LOGF);
    float*    part    = (float*)(ws + WS_PART);
    float*    out     = (float*)d_out;

    bsl_zero_hist<<<1, 1024, 0, stream>>>(hist);
    bsl_hist<<<32, 256, 0, stream>>>(target, hist);
    bsl_freq<<<1, 1024, 0, stream>>>(hist, freqf, logfreq);
    bsl_main<<<MAIN_BLOCKS, MAIN_THREADS, 0, stream>>>(pred, target, freqf, logfreq, part);
    bsl_final<<<1, 1024, 0, stream>>>(part, out);
}